// Model_17411797418177
// MI455X (gfx1250) — compile-verified
//
#include <hip/hip_runtime.h>
#include <cstdint>

// ---------------------------------------------------------------------------
// reshape_and_cache scatter for MI455X (gfx1250).
// Pure bandwidth problem: ~2.1 GB of traffic -> ~90us floor at 23.3 TB/s.
// Phase 1: bulk cache copy, b128 non-temporal loads/stores, x4 unrolled so
//          each wave keeps 4 b128 loads in flight per iteration (clause),
//          launched once per cache tensor (no per-iteration pointer select).
// Phase 2: token scatter via the CDNA5 async global<->LDS DMA path
//          (global_load_async_to_lds_b128 / global_store_async_from_lds_b128,
//          ASYNCcnt-tracked) -- data never round-trips through VGPRs.
// ---------------------------------------------------------------------------

typedef float v4f __attribute__((ext_vector_type(4)));

static constexpr int  NUM_BLOCKS    = 1024;
static constexpr int  NUM_KV_SLICES = 64;
static constexpr int  BLOCK_SIZE    = 128;   // tokens per block
static constexpr int  LAST_DIM      = 16;
static constexpr int  N_TOKENS      = 4096;
static constexpr int  HIDDEN        = NUM_KV_SLICES * LAST_DIM;       // 1024
static constexpr long CACHE_ELEMS   =
    (long)NUM_BLOCKS * NUM_KV_SLICES * BLOCK_SIZE * LAST_DIM;         // 134,217,728
static constexpr long CACHE_CHUNKS  = CACHE_ELEMS / 4;                // 33,554,432 float4

// ---------------------------------------------------------------------------
// Phase 1: stream one cache tensor into its slice of d_out.
// Block tile = 1024 chunks (256 thr x 4 chunks). 8192 blocks x 4 sweeps
// covers 33,554,432 chunks exactly -> no tail, no bounds checks in-loop.
// ---------------------------------------------------------------------------
__global__ void kv_bulk_copy(const float* __restrict__ src_f,
                             float* __restrict__ dst_f) {
    const v4f* __restrict__ src = (const v4f*)src_f;
    v4f* __restrict__ dst = (v4f*)dst_f;

    const long tile_stride = (long)gridDim.x * blockDim.x * 4;
    for (long base = (long)blockIdx.x * blockDim.x * 4 + threadIdx.x;
         base < CACHE_CHUNKS; base += tile_stride) {
        // 4 independent b128 NT loads -> one load clause, 4 in flight.
        v4f a = __builtin_nontemporal_load(src + base);
        v4f b = __builtin_nontemporal_load(src + base + 256);
        v4f c = __builtin_nontemporal_load(src + base + 512);
        v4f d = __builtin_nontemporal_load(src + base + 768);
        __builtin_nontemporal_store(a, dst + base);
        __builtin_nontemporal_store(b, dst + base + 256);
        __builtin_nontemporal_store(c, dst + base + 512);
        __builtin_nontemporal_store(d, dst + base + 768);
    }
}

// ---------------------------------------------------------------------------
// Phase 2: scatter token rows into the caches inside d_out.
// One thread per 16-byte chunk. Chunk space: 2 tensors * 4096 tokens *
// 256 float4-chunks = 2,097,152 threads == 8192 blocks * 256 (exact cover,
// EXEC all-ones as required for the async path).
// Each lane: async DMA 16B global->LDS(private slot), wait, async DMA
// 16B LDS->global at the scattered destination.
// ---------------------------------------------------------------------------
__global__ void kv_scatter_async(const float* __restrict__ key,
                                 const float* __restrict__ value,
                                 const int*   __restrict__ slot_mapping,
                                 float* __restrict__ out) {
    __shared__ __align__(16) float lds[256 * 4];   // 16B slot per thread, 4KB/WG

    const int  tid = threadIdx.x;
    long g = (long)blockIdx.x * blockDim.x + tid;  // global chunk id

    const long chunks_per_tensor = (long)N_TOKENS * (HIDDEN / 4);  // 1,048,576
    const bool is_value = (g >= chunks_per_tensor);
    const long c = is_value ? (g - chunks_per_tensor) : g;

    const int t = (int)(c >> 8);      // token        (c / 256)
    const int q = (int)(c & 255);     // float4 chunk within token row
    const int s = q >> 2;             // kv slice     (q / 4)
    const int p = q & 3;              // float4 within the 16-float slice

    const float* src = (is_value ? value : key) + (long)t * HIDDEN + (long)q * 4;

    const int slot = slot_mapping[t];
    const int bidx = slot >> 7;       // slot / BLOCK_SIZE
    const int boff = slot & 127;      // slot % BLOCK_SIZE

    long dst_idx = ((((long)bidx * NUM_KV_SLICES + s) * BLOCK_SIZE + boff) * LAST_DIM)
                   + (long)p * 4;
    float* dst = out + (is_value ? CACHE_ELEMS : 0) + dst_idx;

    // Per-lane LDS byte offset (low 32 bits of the generic shared pointer).
    const uint32_t lds_off = (uint32_t)(uintptr_t)(void*)&lds[tid * 4];
    const uint64_t gsrc = (uint64_t)(uintptr_t)src;
    const uint64_t gdst = (uint64_t)(uintptr_t)dst;

    // global -> LDS (16B per lane), ASYNCcnt-tracked
    asm volatile("global_load_async_to_lds_b128 %0, %1, off"
                 :: "v"(lds_off), "v"(gsrc) : "memory");
    // LDS writes from async loads may land out of order vs. the store's LDS
    // reads -> drain ASYNCcnt before issuing the store-from-LDS.
    asm volatile("s_wait_asynccnt 0" ::: "memory");
    // LDS -> global at the scattered 16B destination
    asm volatile("global_store_async_from_lds_b128 %0, %1, off"
                 :: "v"(gdst), "v"(lds_off) : "memory");
    // Drain before the wave retires (S_ENDPGM also wait-idles, belt+braces).
    asm volatile("s_wait_asynccnt 0" ::: "memory");
}

// ---------------------------------------------------------------------------
// Launch
// ---------------------------------------------------------------------------
extern "C" void kernel_launch(void* const* d_in, const int* in_sizes, int n_in,
                              void* d_out, int out_size, void* d_ws, size_t ws_size,
                              hipStream_t stream) {
    const float* key         = (const float*)d_in[0];
    const float* key_cache   = (const float*)d_in[1];
    const int*   slot_map    = (const int*)  d_in[2];
    const float* value       = (const float*)d_in[3];
    const float* value_cache = (const float*)d_in[4];
    float*       out         = (float*)d_out;

    (void)in_sizes; (void)n_in; (void)out_size; (void)d_ws; (void)ws_size;

    // Phase 1: stream both caches into d_out ([key_cache | value_cache]).
    kv_bulk_copy<<<8192, 256, 0, stream>>>(key_cache, out);
    kv_bulk_copy<<<8192, 256, 0, stream>>>(value_cache, out + CACHE_ELEMS);

    // Phase 2: 2,097,152 chunks == 8192 WGs x 256 thr, exact cover.
    kv_scatter_async<<<8192, 256, 0, stream>>>(key, value, slot_map, out);
}